// VarianceBasedJND_50500225466961
// MI455X (gfx1250) — compile-verified
//
#include <hip/hip_runtime.h>
#include <stdint.h>

// ---------------------------------------------------------------------------
// JND heat-map blend: gray -> sobel^2 -> clip -> affine -> lerp(imgs, imgs_w)
// Bandwidth-bound (144 MiB @ 23.3 TB/s ~ 6.3 us floor). Uses CDNA5 async
// global->LDS staging for the reflect-padded halo tile.
// ---------------------------------------------------------------------------

#define TILE_H 16
#define TILE_W 64
#define HALO_H (TILE_H + 2)      // 18
#define HALO_W (TILE_W + 2)      // 66
#define LDS_W  68                // padded row stride: rows stay 16B aligned

#if defined(__has_builtin)
#  if __has_builtin(__builtin_amdgcn_global_load_async_to_lds_b32)
#    define ATHENA_ASYNC_BUILTIN 1
#  endif
#  if __has_builtin(__builtin_amdgcn_s_wait_asynccnt)
#    define ATHENA_WAITASYNC_BUILTIN 1
#  endif
#endif

typedef __attribute__((address_space(1))) int GlobInt;   // prints as __device__ int
typedef __attribute__((address_space(3))) int LdsInt;    // LDS int
typedef __attribute__((address_space(3))) void LdsVoid;

__device__ __forceinline__ void async_g2l_b32(const float* gsrc, float* ldst) {
#ifdef ATHENA_ASYNC_BUILTIN
  __builtin_amdgcn_global_load_async_to_lds_b32((GlobInt*)gsrc, (LdsInt*)ldst,
                                                /*offset=*/0, /*cpol=*/0);
#else
  uint32_t loff = (uint32_t)(uintptr_t)(LdsVoid*)ldst;   // LDS byte offset
  uint64_t ga   = (uint64_t)(uintptr_t)gsrc;
  asm volatile("global_load_async_to_lds_b32 %0, %1, off"
               :: "v"(loff), "v"(ga) : "memory");
#endif
}

__device__ __forceinline__ void wait_async_zero() {
#ifdef ATHENA_WAITASYNC_BUILTIN
  __builtin_amdgcn_s_wait_asynccnt(0);
#else
  asm volatile("s_wait_asynccnt 0" ::: "memory");
#endif
}

// jnp.pad 'reflect' with halo of 1: -1 -> 1, n -> n-2
__device__ __forceinline__ int refl(int i, int n) {
  return (i < 0) ? -i : ((i >= n) ? (2 * n - 2 - i) : i);
}

__global__ __launch_bounds__(256)
void jnd_heat_blend_kernel(const float* __restrict__ imgs,
                           const float* __restrict__ imgs_w,
                           const float* __restrict__ grayw,
                           const float* __restrict__ kxp,
                           const float* __restrict__ kyp,
                           float* __restrict__ out,
                           int H, int W) {
  __shared__ __align__(16) float srgb[3 * HALO_H * LDS_W];  // raw channel halo
  __shared__ __align__(16) float sgray[HALO_H * LDS_W];     // 255*gray halo

  const int tid = threadIdx.x;
  const int bx = blockIdx.x, by = blockIdx.y, b = blockIdx.z;
  const int x0 = bx * TILE_W, y0 = by * TILE_H;
  const int plane = H * W;
  const float* imgB = imgs + (b * 3) * plane;

  // ---- Stage 1: async global -> LDS staging of 3-channel reflect halo ------
  const int halo_elems = HALO_H * HALO_W;  // 1188
  for (int idx = tid; idx < 3 * halo_elems; idx += 256) {
    int c   = idx / halo_elems;
    int rem = idx - c * halo_elems;
    int r   = rem / HALO_W;
    int col = rem - r * HALO_W;
    int gr = refl(y0 + r - 1, H);
    int gc = refl(x0 + col - 1, W);
    async_g2l_b32(imgB + c * plane + gr * W + gc,
                  &srgb[(c * HALO_H + r) * LDS_W + col]);
  }
  wait_async_zero();
  __syncthreads();

  // ---- Stage 2: gray = 255 * dot(rgb, gray_w), LDS -> LDS ------------------
  const float w0 = grayw[0] * 255.0f;
  const float w1 = grayw[1] * 255.0f;
  const float w2 = grayw[2] * 255.0f;
  for (int idx = tid; idx < halo_elems; idx += 256) {
    int r   = idx / HALO_W;
    int col = idx - r * HALO_W;
    int o   = r * LDS_W + col;
    float g = fmaf(srgb[o], w0,
              fmaf(srgb[HALO_H * LDS_W + o], w1,
                   srgb[2 * HALO_H * LDS_W + o] * w2));
    sgray[o] = g;
  }
  __syncthreads();

  // ---- Stage 3: Sobel + heat map, 4 pixels per thread ----------------------
  float kx[9], ky[9];
#pragma unroll
  for (int i = 0; i < 9; ++i) { kx[i] = kxp[i]; ky[i] = kyp[i]; }  // uniform -> SGPRs

  const int tx = tid & 15;     // float4 group 0..15
  const int ty = tid >> 4;     // row 0..15
  const int colb = tx * 4;

  float row[3][6];
#pragma unroll
  for (int dr = 0; dr < 3; ++dr) {
    const float* base = &sgray[(ty + dr) * LDS_W + colb];   // 16B aligned
    float4 v4 = *(const float4*)base;                        // ds_load_b128
    float2 v2 = *(const float2*)(base + 4);                  // ds_load_b64
    row[dr][0] = v4.x; row[dr][1] = v4.y; row[dr][2] = v4.z; row[dr][3] = v4.w;
    row[dr][4] = v2.x; row[dr][5] = v2.y;
  }

  float h[4];
#pragma unroll
  for (int j = 0; j < 4; ++j) {
    float gxv = 0.0f, gyv = 0.0f;
#pragma unroll
    for (int dr = 0; dr < 3; ++dr) {
#pragma unroll
      for (int dc = 0; dc < 3; ++dc) {
        float v = row[dr][j + dc];
        gxv = fmaf(kx[dr * 3 + dc], v, gxv);
        gyv = fmaf(ky[dr * 3 + dc], v, gyv);
      }
    }
    float cm = fmaf(gxv, gxv, gyv * gyv);
    cm = fminf(fmaxf(cm, 0.0f), 2500.0f);                    // clip
    h[j] = fmaf(cm, (1.0f - 0.2f) / 2500.0f, 0.2f);          // heat map
  }

  // ---- Stage 4: out = a + h*(w - a); center 'a' comes from staged LDS ------
  const int pix = (y0 + ty) * W + (x0 + colb);
#pragma unroll
  for (int c = 0; c < 3; ++c) {
    int off = (b * 3 + c) * plane + pix;
    float4 wv = *(const float4*)(imgs_w + off);              // global_load_b128
    const float* a_lds = &srgb[(c * HALO_H + (ty + 1)) * LDS_W + colb + 1];
    float4 o;
    { float a = a_lds[0]; o.x = fmaf(h[0], wv.x - a, a); }
    { float a = a_lds[1]; o.y = fmaf(h[1], wv.y - a, a); }
    { float a = a_lds[2]; o.z = fmaf(h[2], wv.z - a, a); }
    { float a = a_lds[3]; o.w = fmaf(h[3], wv.w - a, a); }
    *(float4*)(out + off) = o;                                // global_store_b128
  }
}

extern "C" void kernel_launch(void* const* d_in, const int* in_sizes, int n_in,
                              void* d_out, int out_size, void* d_ws, size_t ws_size,
                              hipStream_t stream) {
  const float* imgs   = (const float*)d_in[0];
  const float* imgs_w = (const float*)d_in[1];
  const float* grayw  = (const float*)d_in[2];
  const float* kx     = (const float*)d_in[3];
  const float* ky     = (const float*)d_in[4];
  float* out = (float*)d_out;

  const int H = 512, W = 512;
  const int B = in_sizes[0] / (3 * H * W);   // 16

  dim3 grid(W / TILE_W, H / TILE_H, B);      // (8, 32, 16)
  jnd_heat_blend_kernel<<<grid, 256, 0, stream>>>(imgs, imgs_w, grayw, kx, ky,
                                                  out, H, W);
}